// GCN_75050258530542
// MI455X (gfx1250) — compile-verified
//
#include <hip/hip_runtime.h>

#define N_NODES  100000
#define N_EDGES  3200000
#define F_IN     10      // feature slice width
#define KPAD     12      // K padded to multiple of 4 for 16x16x4 WMMA
#define N_OUT    64

typedef __attribute__((ext_vector_type(2))) float v2f;
typedef __attribute__((ext_vector_type(8))) float v8f;

// ---------------- zero scratch ----------------
__global__ void __launch_bounds__(256) gcn_zero_kernel(float* p, int n) {
    int i = blockIdx.x * blockDim.x + threadIdx.x;
    if (i < n) p[i] = 0.0f;
}

// ---------------- degree counts (float atomics, L2-resident) ----------------
__global__ void __launch_bounds__(256) gcn_degree_kernel(const int* __restrict__ src,
                                                         const int* __restrict__ dst,
                                                         float* __restrict__ deg_out,
                                                         float* __restrict__ deg_in,
                                                         int nE) {
    int e = blockIdx.x * blockDim.x + threadIdx.x;
    if (e < nE) {
        atomicAdd(&deg_out[src[e]], 1.0f);
        atomicAdd(&deg_in [dst[e]], 1.0f);
    }
}

// ---------------- per-node: norms + pre-scaled feature slice ----------------
__global__ void __launch_bounds__(256) gcn_nodeprep_kernel(const float* __restrict__ x,
                                                           const float* __restrict__ deg_out,
                                                           const float* __restrict__ deg_in,
                                                           float* __restrict__ norm_d,
                                                           float* __restrict__ xscaled,
                                                           int nN) {
    int n = blockIdx.x * blockDim.x + threadIdx.x;
    if (n < nN) {
        float dout = deg_out[n]; dout = (dout > 0.0f) ? dout : 1.0f;
        float din  = deg_in [n]; din  = (din  > 0.0f) ? din  : 1.0f;
        float ns = rsqrtf(dout);
        norm_d[n] = rsqrtf(din);
        const float* xr = x + (size_t)n * 40 + 15;
        float* o = xscaled + (size_t)n * F_IN;
#pragma unroll
        for (int k = 0; k < F_IN; ++k) o[k] = xr[k] * ns;
    }
}

// ---------------- fold W1@W2 -> Wc (KPAD x 64, pad rows zero), bc = b1@W2+b2 ----------------
__global__ void __launch_bounds__(64) gcn_foldw_kernel(const float* __restrict__ W1,
                                                       const float* __restrict__ b1,
                                                       const float* __restrict__ W2,
                                                       const float* __restrict__ b2,
                                                       float* __restrict__ Wc,
                                                       float* __restrict__ bc) {
    int j = threadIdx.x;            // 0..63 output column
    float acc = b2[j];
    for (int k = 0; k < 128; ++k) acc += b1[k] * W2[k * 64 + j];
    bc[j] = acc;
#pragma unroll 1
    for (int i = 0; i < KPAD; ++i) {
        float s = 0.0f;
        if (i < F_IN)
            for (int k = 0; k < 128; ++k) s += W1[i * 128 + k] * W2[k * 64 + j];
        Wc[i * 64 + j] = s;
    }
}

// ---------------- edge scatter: agg[dst] += xscaled[src] ----------------
__global__ void __launch_bounds__(256) gcn_scatter_kernel(const int* __restrict__ src,
                                                          const int* __restrict__ dst,
                                                          const float* __restrict__ xscaled,
                                                          float* __restrict__ agg,
                                                          int nE) {
    int e = blockIdx.x * blockDim.x + threadIdx.x;
    if (e < nE) {
        int s = src[e], d = dst[e];
        const float* xs = xscaled + (size_t)s * F_IN;
        float*       a  = agg     + (size_t)d * F_IN;
        float v[F_IN];
#pragma unroll
        for (int k = 0; k < F_IN; ++k) v[k] = xs[k];
#pragma unroll
        for (int k = 0; k < F_IN; ++k) atomicAdd(&a[k], v[k]);
    }
}

// ---------------- out = (agg * norm_d) @ Wc + bc via V_WMMA_F32_16X16X4_F32 ----------------
// One wave (32 lanes) per 16-node tile; N=64 -> 4 n-tiles; K=12 -> 3 k-steps.
__global__ void __launch_bounds__(32) gcn_wmma_kernel(const float* __restrict__ agg,
                                                      const float* __restrict__ norm_d,
                                                      const float* __restrict__ Wc,
                                                      const float* __restrict__ bc,
                                                      float* __restrict__ out) {
    int tile = blockIdx.x;          // 100000/16 = 6250 tiles, exact
    int lane = threadIdx.x;
    int m  = lane & 15;             // A-matrix row (M)
    int kh = lane >> 4;             // lane half selects K pair (ISA 16x4 A layout)

    int node = tile * 16 + m;
    float nd = norm_d[node];
    const float* arow = agg + (size_t)node * F_IN;

    // A fragments for k0 = 0,4,8.  VGPR0: K=k0+2*kh ; VGPR1: K=k0+2*kh+1
    v2f afrag[3];
#pragma unroll
    for (int ks = 0; ks < 3; ++ks) {
        int k0 = ks * 4 + kh * 2;
        afrag[ks].x = (k0     < F_IN) ? arow[k0]     * nd : 0.0f;
        afrag[ks].y = (k0 + 1 < F_IN) ? arow[k0 + 1] * nd : 0.0f;
    }

    int nn = lane & 15;             // B/C/D column within n-tile
#pragma unroll
    for (int j = 0; j < 4; ++j) {   // n-tiles: columns [16j, 16j+16)
        v8f c = {};
#pragma unroll
        for (int ks = 0; ks < 3; ++ks) {
            int kb = ks * 4 + kh * 2;   // B 4x16 layout mirrors A: lane half picks K pair
            v2f b;
            b.x = Wc[(kb    ) * N_OUT + j * 16 + nn];
            b.y = Wc[(kb + 1) * N_OUT + j * 16 + nn];
            c = __builtin_amdgcn_wmma_f32_16x16x4_f32(
                    /*neg_a=*/false, afrag[ks],
                    /*neg_b=*/false, b,
                    /*c_mod=*/(short)0, c,
                    /*reuse_a=*/false, /*reuse_b=*/false);
        }
        float bias = bc[j * 16 + nn];
        int nbase = tile * 16;
#pragma unroll
        for (int r = 0; r < 8; ++r) {   // VGPR r holds rows M=r (lanes 0-15) / M=r+8 (16-31)
            int row = r + 8 * kh;
            out[(size_t)(nbase + row) * N_OUT + j * 16 + nn] = c[r] + bias;
        }
    }
}

extern "C" void kernel_launch(void* const* d_in, const int* in_sizes, int n_in,
                              void* d_out, int out_size, void* d_ws, size_t ws_size,
                              hipStream_t stream) {
    const float* x   = (const float*)d_in[0];
    const int*   src = (const int*)  d_in[1];
    const int*   dst = (const int*)  d_in[2];
    const float* W1  = (const float*)d_in[3];
    const float* b1  = (const float*)d_in[4];
    const float* W2  = (const float*)d_in[5];
    const float* b2  = (const float*)d_in[6];
    float* out = (float*)d_out;

    float* ws = (float*)d_ws;
    // layout: zeroed block first (deg_out | deg_in | agg), then non-zeroed scratch
    float* deg_out = ws;                                  // 100000
    float* deg_in  = ws + N_NODES;                        // 100000
    float* agg     = ws + 2 * N_NODES;                    // 100000*10
    float* norm_dv = ws + 2 * N_NODES + N_NODES * F_IN;   // 100000
    float* xscaled = norm_dv + N_NODES;                   // 100000*10
    float* Wc      = xscaled + (size_t)N_NODES * F_IN;    // 12*64
    float* bcv     = Wc + KPAD * N_OUT;                   // 64

    const int zeroN = 2 * N_NODES + N_NODES * F_IN;       // deg_out+deg_in+agg
    gcn_zero_kernel<<<(zeroN + 255) / 256, 256, 0, stream>>>(ws, zeroN);

    gcn_degree_kernel<<<(N_EDGES + 255) / 256, 256, 0, stream>>>(src, dst, deg_out, deg_in, N_EDGES);

    gcn_nodeprep_kernel<<<(N_NODES + 255) / 256, 256, 0, stream>>>(x, deg_out, deg_in,
                                                                   norm_dv, xscaled, N_NODES);

    gcn_foldw_kernel<<<1, 64, 0, stream>>>(W1, b1, W2, b2, Wc, bcv);

    gcn_scatter_kernel<<<(N_EDGES + 255) / 256, 256, 0, stream>>>(src, dst, xscaled, agg, N_EDGES);

    gcn_wmma_kernel<<<N_NODES / 16, 32, 0, stream>>>(agg, norm_dv, Wc, bcv, out);
}